// FeatGuide_BatchDrop_66606352827269
// MI455X (gfx1250) — compile-verified
//
#include <hip/hip_runtime.h>
#include <math.h>

typedef __attribute__((ext_vector_type(2))) float v2f;
typedef __attribute__((ext_vector_type(8))) float v8f;

// ---------------------------------------------------------------------------
// Problem constants (fixed by setup_inputs)
// ---------------------------------------------------------------------------
#define B_    32
#define C_    256
#define H_    64
#define W_    64
#define HW_   (H_ * W_)
#define RSE_  16      // c / 16
#define RDY_  64      // c / 4
#define RH_   3       // int(0.05 * 64)
#define RW_   6       // int(0.10 * 64)

__device__ __forceinline__ float sigm(float x) {
    return 1.0f / (1.0f + expf(-x));
}

// ---------------------------------------------------------------------------
// Core WMMA helper: one wave accumulates a 16x16 f32 tile of  A(16xK) * W^T
// using V_WMMA_F32_16X16X4_F32.  arow / wrow are per-lane base pointers
// already offset by row and by the lane-half K offset (+0 or +2).
// ---------------------------------------------------------------------------
__device__ __forceinline__ v8f wmma_mac(const float* __restrict__ arow,
                                        const float* __restrict__ wrow,
                                        int K, v8f acc) {
    for (int k = 0; k < K; k += 4) {
        v2f a = *(const v2f*)(arow + k);
        v2f b = *(const v2f*)(wrow + k);
        acc = __builtin_amdgcn_wmma_f32_16x16x4_f32(
            /*neg_a=*/false, a, /*neg_b=*/false, b,
            /*c_mod=*/(short)0, acc, /*reuse_a=*/false, /*reuse_b=*/false);
    }
    return acc;
}

__device__ __forceinline__ void store_tile(float* __restrict__ D, int ldd,
                                           int m0, int n0, v8f acc) {
    const int lane = threadIdx.x & 31;
    const int half = lane >> 4;
    const int l    = lane & 15;
    #pragma unroll
    for (int r = 0; r < 8; ++r)
        D[(size_t)(m0 + r + 8 * half) * ldd + n0 + l] = acc[r];
}

// ---------------------------------------------------------------------------
// K1: per-(b,c) spatial mean of x.  8192 blocks x 256 threads, float4 loads.
// ---------------------------------------------------------------------------
__global__ void mean_kernel(const float* __restrict__ x, float* __restrict__ m) {
    __shared__ float red[256];
    const int plane = blockIdx.x;                     // b*256 + c
    const float4* px = (const float4*)(x + (size_t)plane * HW_);
    float s = 0.0f;
    #pragma unroll
    for (int i = 0; i < 4; ++i) {
        float4 v = px[threadIdx.x + 256 * i];
        s += v.x + v.y + v.z + v.w;
    }
    red[threadIdx.x] = s;
    __syncthreads();
    for (int off = 128; off > 0; off >>= 1) {
        if (threadIdx.x < off) red[threadIdx.x] += red[threadIdx.x + off];
        __syncthreads();
    }
    if (threadIdx.x == 0) m[plane] = red[0] * (1.0f / (float)HW_);
}

// ---------------------------------------------------------------------------
// K2: t1 = relu(m @ se_w1^T)   M=32 N=16 K=256  -> 2 tiles (1 block, 2 waves)
// ---------------------------------------------------------------------------
__global__ void se1_kernel(const float* __restrict__ m,
                           const float* __restrict__ w1,
                           float* __restrict__ t1) {
    const int wave = threadIdx.x >> 5;
    const int lane = threadIdx.x & 31, half = lane >> 4, l = lane & 15;
    const int m0 = wave * 16;
    v8f acc = {0.f, 0.f, 0.f, 0.f, 0.f, 0.f, 0.f, 0.f};
    acc = wmma_mac(m + (size_t)(m0 + l) * C_ + 2 * half,
                   w1 + (size_t)l * C_ + 2 * half, C_, acc);
    #pragma unroll
    for (int r = 0; r < 8; ++r)
        t1[(size_t)(m0 + r + 8 * half) * RSE_ + l] = fmaxf(acc[r], 0.0f);
}

// ---------------------------------------------------------------------------
// K3: ysc = sigmoid(t1 @ se_w2^T);  g = m * ysc.   M=32 N=256 K=16 -> 32 tiles
// ---------------------------------------------------------------------------
__global__ void se2_kernel(const float* __restrict__ t1,
                           const float* __restrict__ w2,
                           const float* __restrict__ m,
                           float* __restrict__ ysc,
                           float* __restrict__ g) {
    const int wave = (blockIdx.x * blockDim.x + threadIdx.x) >> 5;
    const int lane = threadIdx.x & 31, half = lane >> 4, l = lane & 15;
    const int m0 = (wave & 1) * 16, n0 = (wave >> 1) * 16;
    v8f acc = {0.f, 0.f, 0.f, 0.f, 0.f, 0.f, 0.f, 0.f};
    acc = wmma_mac(t1 + (size_t)(m0 + l) * RSE_ + 2 * half,
                   w2 + (size_t)(n0 + l) * RSE_ + 2 * half, RSE_, acc);
    #pragma unroll
    for (int r = 0; r < 8; ++r) {
        const int row = m0 + r + 8 * half, col = n0 + l;
        const float y = sigm(acc[r]);
        ysc[(size_t)row * C_ + col] = y;
        g[(size_t)row * C_ + col]   = m[(size_t)row * C_ + col] * y;
    }
}

// ---------------------------------------------------------------------------
// K4: vh = g @ ch_w^T + ch_b ; vw = g @ cw_w^T + cw_b.
//     Each M=32 N=256 K=256 -> 32 tiles;  64 waves total (8 blocks x 8 waves).
// ---------------------------------------------------------------------------
__global__ void chan_kernel(const float* __restrict__ g,
                            const float* __restrict__ chw, const float* __restrict__ chb,
                            const float* __restrict__ cww, const float* __restrict__ cwb,
                            float* __restrict__ vh, float* __restrict__ vw) {
    const int wave = (blockIdx.x * blockDim.x + threadIdx.x) >> 5;
    const int lane = threadIdx.x & 31, half = lane >> 4, l = lane & 15;
    const int path = wave >> 5, t = wave & 31;
    const int m0 = (t & 1) * 16, n0 = (t >> 1) * 16;
    const float* Wm = path ? cww : chw;
    const float* Bv = path ? cwb : chb;
    float* O        = path ? vw  : vh;
    v8f acc;
    const float bj = Bv[n0 + l];
    #pragma unroll
    for (int r = 0; r < 8; ++r) acc[r] = bj;
    acc = wmma_mac(g  + (size_t)(m0 + l) * C_ + 2 * half,
                   Wm + (size_t)(n0 + l) * C_ + 2 * half, C_, acc);
    store_tile(O, C_, m0, n0, acc);
}

// ---------------------------------------------------------------------------
// K5: th = relu(v @ fc1w^T + fc1b).  M=32 N=64 K=256 -> 8 tiles/path,
//     16 waves (1 block x 512 threads).
// ---------------------------------------------------------------------------
__global__ void fc1_kernel(const float* __restrict__ vh, const float* __restrict__ vw,
                           const float* __restrict__ w1h, const float* __restrict__ b1h,
                           const float* __restrict__ w1w, const float* __restrict__ b1w,
                           float* __restrict__ thh, float* __restrict__ thw) {
    const int wave = threadIdx.x >> 5;
    const int lane = threadIdx.x & 31, half = lane >> 4, l = lane & 15;
    const int path = wave >> 3, t = wave & 7;
    const int m0 = (t & 1) * 16, n0 = (t >> 1) * 16;
    const float* V  = path ? vw  : vh;
    const float* Wm = path ? w1w : w1h;
    const float* Bv = path ? b1w : b1h;
    float* O        = path ? thw : thh;
    v8f acc;
    const float bj = Bv[n0 + l];
    #pragma unroll
    for (int r = 0; r < 8; ++r) acc[r] = bj;
    acc = wmma_mac(V  + (size_t)(m0 + l) * C_ + 2 * half,
                   Wm + (size_t)(n0 + l) * C_ + 2 * half, C_, acc);
    #pragma unroll
    for (int r = 0; r < 8; ++r)
        O[(size_t)(m0 + r + 8 * half) * RDY_ + n0 + l] = fmaxf(acc[r], 0.0f);
}

// ---------------------------------------------------------------------------
// K6: dyrelu tail.  For each (b,c) tile, accumulate the 4 interleaved columns
//     of  t2 = th @ fc2w^T + fc2b  (rows 4c+j of fc2w, K=64) via WMMA, then
//     fold into sx/sy without materializing t2.
//     64 waves (8 blocks x 8 waves), 2 paths x 32 tiles.
// ---------------------------------------------------------------------------
__global__ void fc2_kernel(const float* __restrict__ thh, const float* __restrict__ thw,
                           const float* __restrict__ w2h, const float* __restrict__ b2h,
                           const float* __restrict__ w2w, const float* __restrict__ b2w,
                           const float* __restrict__ vh,  const float* __restrict__ vw,
                           float* __restrict__ sx, float* __restrict__ sy) {
    const int wave = (blockIdx.x * blockDim.x + threadIdx.x) >> 5;
    const int lane = threadIdx.x & 31, half = lane >> 4, l = lane & 15;
    const int path = wave >> 5, t = wave & 31;
    const int m0 = (t & 1) * 16, n0 = (t >> 1) * 16;
    const float* TH = path ? thw : thh;
    const float* W2 = path ? w2w : w2h;
    const float* B2 = path ? b2w : b2h;
    const float* V  = path ? vw  : vh;
    float* OUT      = path ? sy  : sx;
    const float lim = path ? (float)(W_ - RW_) : (float)(H_ - RH_);

    const float* arow = TH + (size_t)(m0 + l) * RDY_ + 2 * half;
    v8f acc[4];
    #pragma unroll
    for (int j = 0; j < 4; ++j) {
        const int wr = 4 * (n0 + l) + j;
        const float bj = B2[wr];
        v8f c;
        #pragma unroll
        for (int r = 0; r < 8; ++r) c[r] = bj;
        acc[j] = wmma_mac(arow, W2 + (size_t)wr * RDY_ + 2 * half, RDY_, c);
    }
    #pragma unroll
    for (int r = 0; r < 8; ++r) {
        const int row = m0 + r + 8 * half, col = n0 + l;
        const float v   = V[(size_t)row * C_ + col];
        const float th0 = 2.0f * sigm(acc[0][r]) - 1.0f;   // * lambda 1.0 + init 1.0
        const float th1 = 2.0f * sigm(acc[1][r]) - 1.0f;   // * lambda 1.0 + init 0.0
        const float th2 = 2.0f * sigm(acc[2][r]) - 1.0f;   // * lambda 0.5
        const float th3 = 2.0f * sigm(acc[3][r]) - 1.0f;   // * lambda 0.5
        const float dy  = fmaxf(v * (th0 + 1.0f) + 0.5f * th2,
                                v * th1          + 0.5f * th3);
        const float s   = ceilf((float)H_ * sigm(dy));     // ref uses h for both
        OUT[(size_t)row * C_ + col] = fminf(s, lim);
    }
}

// ---------------------------------------------------------------------------
// K7: out = x * ysc * !(inside 3x6 drop block).  Streaming, float4.
// ---------------------------------------------------------------------------
__global__ void apply_kernel(const float* __restrict__ x,
                             const float* __restrict__ ysc,
                             const float* __restrict__ sx,
                             const float* __restrict__ sy,
                             float* __restrict__ out) {
    const int plane = blockIdx.x;                   // b*256 + c
    const float sc = ysc[plane];
    const float x0 = sx[plane];
    const float y0 = sy[plane];
    const float4* px = (const float4*)(x + (size_t)plane * HW_);
    float4* po = (float4*)(out + (size_t)plane * HW_);
    #pragma unroll
    for (int i = 0; i < 4; ++i) {
        const int idx4 = threadIdx.x + 256 * i;
        float4 v = px[idx4];
        const int f = idx4 * 4;
        const float r = (float)(f >> 6);
        const float c = (float)(f & (W_ - 1));
        const bool rin = (r >= x0) && (r < x0 + (float)RH_);
        float4 o;
        const bool i0 = rin && (c          >= y0) && (c          < y0 + (float)RW_);
        const bool i1 = rin && (c + 1.0f   >= y0) && (c + 1.0f   < y0 + (float)RW_);
        const bool i2 = rin && (c + 2.0f   >= y0) && (c + 2.0f   < y0 + (float)RW_);
        const bool i3 = rin && (c + 3.0f   >= y0) && (c + 3.0f   < y0 + (float)RW_);
        o.x = i0 ? 0.0f : v.x * sc;
        o.y = i1 ? 0.0f : v.y * sc;
        o.z = i2 ? 0.0f : v.z * sc;
        o.w = i3 ? 0.0f : v.w * sc;
        po[idx4] = o;
    }
}

// ---------------------------------------------------------------------------
// Launch
// ---------------------------------------------------------------------------
extern "C" void kernel_launch(void* const* d_in, const int* in_sizes, int n_in,
                              void* d_out, int out_size, void* d_ws, size_t ws_size,
                              hipStream_t stream) {
    const float* x       = (const float*)d_in[0];
    const float* se_w1   = (const float*)d_in[1];
    const float* se_w2   = (const float*)d_in[2];
    const float* ch_w    = (const float*)d_in[3];
    const float* ch_b    = (const float*)d_in[4];
    const float* cw_w    = (const float*)d_in[5];
    const float* cw_b    = (const float*)d_in[6];
    const float* dh_fc1w = (const float*)d_in[7];
    const float* dh_fc1b = (const float*)d_in[8];
    const float* dh_fc2w = (const float*)d_in[9];
    const float* dh_fc2b = (const float*)d_in[10];
    const float* dw_fc1w = (const float*)d_in[11];
    const float* dw_fc1b = (const float*)d_in[12];
    const float* dw_fc2w = (const float*)d_in[13];
    const float* dw_fc2b = (const float*)d_in[14];
    float* out = (float*)d_out;

    float* ws  = (float*)d_ws;
    float* m   = ws;                 // 8192
    float* t1  = m   + B_ * C_;      // 512
    float* ysc = t1  + B_ * RSE_;    // 8192
    float* g   = ysc + B_ * C_;      // 8192
    float* vh  = g   + B_ * C_;      // 8192
    float* vw  = vh  + B_ * C_;      // 8192
    float* thh = vw  + B_ * C_;      // 2048
    float* thw = thh + B_ * RDY_;    // 2048
    float* sx  = thw + B_ * RDY_;    // 8192
    float* sy  = sx  + B_ * C_;      // 8192   (total ~242 KB)

    const int nplanes = B_ * C_;     // 8192

    mean_kernel<<<nplanes, 256, 0, stream>>>(x, m);
    se1_kernel <<<1,   64, 0, stream>>>(m, se_w1, t1);
    se2_kernel <<<4,  256, 0, stream>>>(t1, se_w2, m, ysc, g);
    chan_kernel<<<8,  256, 0, stream>>>(g, ch_w, ch_b, cw_w, cw_b, vh, vw);
    fc1_kernel <<<1,  512, 0, stream>>>(vh, vw, dh_fc1w, dh_fc1b, dw_fc1w, dw_fc1b, thh, thw);
    fc2_kernel <<<8,  256, 0, stream>>>(thh, thw, dh_fc2w, dh_fc2b, dw_fc2w, dw_fc2b,
                                        vh, vw, sx, sy);
    apply_kernel<<<nplanes, 256, 0, stream>>>(x, ysc, sx, sy, out);
}